// Loss_gr_sim_88012469829943
// MI455X (gfx1250) — compile-verified
//
#include <hip/hip_runtime.h>

typedef __attribute__((ext_vector_type(2))) float v2f;
typedef __attribute__((ext_vector_type(8))) float v8f;

#define KCLS  1000
#define DDIM  2048
#define BATCH 8192
#define TKD   4.0f
#define KT    63                 // ceil(1000/16)
#define NTILES (KT * KT)         // 3969 16x16 output tiles of G

// ---------------------------------------------------------------------------
// Kernel 0: zero the scalar output (harness poisons d_out with 0xAA).
// ---------------------------------------------------------------------------
__global__ void init_out(float* out) {
    if (threadIdx.x == 0) out[0] = 0.0f;
}

// ---------------------------------------------------------------------------
// Kernel 1: G = W * W^T  (K x K), full f32 via V_WMMA_F32_16X16X4_F32.
// One wave32 per 16x16 tile; 8 waves per block. EXEC stays all-1s inside a
// wave (tile predicate is wave-uniform), as WMMA requires.
//
// A-matrix 16x4 f32 layout (ISA 7.12.2): lanes 0-15 hold M=lane, lanes 16-31
// hold M=lane-16; VGPR0 = K={0|2}, VGPR1 = K={1|3} depending on lane half.
// -> per lane: contiguous float2 at column offset koff = (lane>=16)?2:0.
// B mirrors this with N in place of M (G is symmetric so both operands are
// plain W rows).
// C/D 16x16 f32: VGPR i -> M = i + 8*(lane>=16), N = lane&15.
// ---------------------------------------------------------------------------
__global__ __launch_bounds__(256) void gram_wmma(const float* __restrict__ W,
                                                 float* __restrict__ G) {
    const int warp = threadIdx.x >> 5;
    const int lane = threadIdx.x & 31;
    const int t = blockIdx.x * 8 + warp;
    if (t >= NTILES) return;                       // wave-uniform exit

    const int tm = t / KT, tn = t % KT;
    const int m0 = tm * 16, n0 = tn * 16;

    int rA = m0 + (lane & 15); if (rA >= KCLS) rA = KCLS - 1;  // clamp; rows
    int rB = n0 + (lane & 15); if (rB >= KCLS) rB = KCLS - 1;  // >=K not stored
    const int koff = (lane >> 4) * 2;

    const float* pa = W + (size_t)rA * DDIM + koff;
    const float* pb = W + (size_t)rB * DDIM + koff;

    v8f c = {};
    for (int kk = 0; kk < DDIM; kk += 16) {
        // speculative prefetch 256B ahead on both operand streams
        __builtin_prefetch(pa + kk + 64, 0, 0);
        __builtin_prefetch(pb + kk + 64, 0, 0);

        v2f a0 = *(const v2f*)(pa + kk + 0);
        v2f b0 = *(const v2f*)(pb + kk + 0);
        v2f a1 = *(const v2f*)(pa + kk + 4);
        v2f b1 = *(const v2f*)(pb + kk + 4);
        v2f a2 = *(const v2f*)(pa + kk + 8);
        v2f b2 = *(const v2f*)(pb + kk + 8);
        v2f a3 = *(const v2f*)(pa + kk + 12);
        v2f b3 = *(const v2f*)(pb + kk + 12);

        c = __builtin_amdgcn_wmma_f32_16x16x4_f32(false, a0, false, b0, (short)0, c, false, false);
        c = __builtin_amdgcn_wmma_f32_16x16x4_f32(false, a1, false, b1, (short)0, c, false, false);
        c = __builtin_amdgcn_wmma_f32_16x16x4_f32(false, a2, false, b2, (short)0, c, false, false);
        c = __builtin_amdgcn_wmma_f32_16x16x4_f32(false, a3, false, b3, (short)0, c, false, false);
    }

    const int n = n0 + (lane & 15);
    const int mbase = m0 + ((lane >> 4) * 8);
    if (n < KCLS) {
        for (int i = 0; i < 8; ++i) {
            const int m = mbase + i;
            if (m < KCLS) G[(size_t)m * KCLS + n] = c[i];
        }
    }
}

// ---------------------------------------------------------------------------
// Block reductions (wave32 shuffles + LDS across the 8 waves of a block).
// ---------------------------------------------------------------------------
__device__ __forceinline__ float block_sum(float v, float* red) {
    const int lane = threadIdx.x & 31, warp = threadIdx.x >> 5;
    #pragma unroll
    for (int o = 16; o > 0; o >>= 1) v += __shfl_xor(v, o, 32);
    __syncthreads();                 // guard red reuse across calls
    if (lane == 0) red[warp] = v;
    __syncthreads();
    float tot = 0.0f;
    #pragma unroll
    for (int w = 0; w < 8; ++w) tot += red[w];
    return tot;
}

__device__ __forceinline__ float block_max(float v, float* red) {
    const int lane = threadIdx.x & 31, warp = threadIdx.x >> 5;
    #pragma unroll
    for (int o = 16; o > 0; o >>= 1) v = fmaxf(v, __shfl_xor(v, o, 32));
    __syncthreads();
    if (lane == 0) red[warp] = v;
    __syncthreads();
    float tot = red[0];
    #pragma unroll
    for (int w = 1; w < 8; ++w) tot = fmaxf(tot, red[w]);
    return tot;
}

// ---------------------------------------------------------------------------
// Kernel 2: in-place per-row  S[l,:] = softmax( relu(G[l,:])^0.3 / 0.3 ).
// One 256-thread block per row l (1000 rows).
// ---------------------------------------------------------------------------
__global__ __launch_bounds__(256) void rowsoftmax(float* __restrict__ G) {
    __shared__ float red[8];
    const int l = blockIdx.x;
    float* row = G + (size_t)l * KCLS;
    const int tid = threadIdx.x;

    float xs[4];
    float mx = -1e30f;
    #pragma unroll
    for (int i = 0; i < 4; ++i) {
        const int k = tid + i * 256;
        float x = -1e30f;
        if (k < KCLS) {
            float g = fmaxf(row[k], 0.0f);
            x = __powf(g, 0.3f) * (1.0f / 0.3f);   // relu(g)^0.3 / 0.3
        }
        xs[i] = x;
        mx = fmaxf(mx, x);
    }
    const float m = block_max(mx, red);

    float s = 0.0f;
    #pragma unroll
    for (int i = 0; i < 4; ++i) {
        const int k = tid + i * 256;
        float e = (k < KCLS) ? __expf(xs[i] - m) : 0.0f;
        xs[i] = e;
        s += e;
    }
    const float inv = 1.0f / block_sum(s, red);

    #pragma unroll
    for (int i = 0; i < 4; ++i) {
        const int k = tid + i * 256;
        if (k < KCLS) row[k] = xs[i] * inv;
    }
}

// ---------------------------------------------------------------------------
// Kernel 3: per-batch-row loss. One 256-thread block per b (8192 rows).
//   predict = pred/T - lse(pred/T);  sum_k target = 1 exactly, so
//   loss_b = H - A1 + lse_p, with H = sum target*log(target),
//   A1 = sum target*(pred/T).
// ---------------------------------------------------------------------------
__global__ __launch_bounds__(256) void loss_kernel(const float* __restrict__ pred,
                                                   const float* __restrict__ teacher,
                                                   const float* __restrict__ S,
                                                   const int* __restrict__ label,
                                                   float* __restrict__ out) {
    __shared__ float red[8];
    __shared__ float conf_sh;

    const int b = blockIdx.x;
    const int l = label[b];
    const float* pr = pred + (size_t)b * KCLS;
    const float* te = teacher + (size_t)b * KCLS;
    const float* srow = S + (size_t)l * KCLS;
    const int tid = threadIdx.x;

    float ps[4], ts[4];
    float mp = -1e30f, mt = -1e30f;
    #pragma unroll
    for (int i = 0; i < 4; ++i) {
        const int k = tid + i * 256;
        float p = -1e30f, t = -1e30f;
        if (k < KCLS) {
            p = pr[k] * (1.0f / TKD);
            t = te[k] * (1.0f / TKD);
        }
        ps[i] = p; ts[i] = t;
        mp = fmaxf(mp, p);
        mt = fmaxf(mt, t);
    }
    mp = block_max(mp, red);
    mt = block_max(mt, red);

    float zp = 0.0f, zt = 0.0f;
    #pragma unroll
    for (int i = 0; i < 4; ++i) {
        zp += __expf(ps[i] - mp);   // exp(-huge) -> 0 for padded lanes
        zt += __expf(ts[i] - mt);
    }
    zp = block_sum(zp, red);
    zt = block_sum(zt, red);
    const float lse_p = mp + __logf(zp);

    // confidence of the true class: owned by exactly one thread
    #pragma unroll
    for (int i = 0; i < 4; ++i) {
        const int k = tid + i * 256;
        if (k == l) conf_sh = __expf(ts[i] - mt) / zt;
    }
    __syncthreads();
    const float conf = conf_sh;
    const float u = (1.0f - conf) * (1.0f / (float)(KCLS - 1));

    float A1 = 0.0f, H = 0.0f;
    #pragma unroll
    for (int i = 0; i < 4; ++i) {
        const int k = tid + i * 256;
        if (k < KCLS) {
            const float gr  = (k == l) ? conf : u;
            const float tgt = 0.5f * srow[k] + 0.5f * gr;
            A1 += tgt * ps[i];
            H  += tgt * __logf(tgt);
        }
    }
    A1 = block_sum(A1, red);
    H  = block_sum(H, red);

    if (tid == 0) {
        const float loss_b = H - A1 + lse_p;
        atomicAdd(out, (TKD * TKD / (float)BATCH) * loss_b);
    }
}

// ---------------------------------------------------------------------------
extern "C" void kernel_launch(void* const* d_in, const int* in_sizes, int n_in,
                              void* d_out, int out_size, void* d_ws, size_t ws_size,
                              hipStream_t stream) {
    const float* pred    = (const float*)d_in[0];   // [B,K] f32
    const float* teacher = (const float*)d_in[1];   // [B,K] f32
    const float* weight  = (const float*)d_in[2];   // [K,D] f32
    const int*   label   = (const int*)d_in[3];     // [B]   int32
    float* out = (float*)d_out;                     // scalar f32
    float* G   = (float*)d_ws;                      // K*K f32 = 4 MB scratch

    init_out<<<1, 32, 0, stream>>>(out);
    gram_wmma<<<(NTILES + 7) / 8, 256, 0, stream>>>(weight, G);
    rowsoftmax<<<KCLS, 256, 0, stream>>>(G);
    loss_kernel<<<BATCH, 256, 0, stream>>>(pred, teacher, G, label, out);
}